// IERGCN_50199577756294
// MI455X (gfx1250) — compile-verified
//
#include <hip/hip_runtime.h>
#include <cmath>

typedef float v2f __attribute__((ext_vector_type(2)));
typedef float v8f __attribute__((ext_vector_type(8)));
typedef int   v4i __attribute__((vector_size(16)));

#define GLOBAL_AS __attribute__((address_space(1)))
#define LDS_AS    __attribute__((address_space(3)))

#if defined(__AMDGCN__) && __has_builtin(__builtin_amdgcn_global_load_async_to_lds_b128) && \
    __has_builtin(__builtin_amdgcn_s_wait_asynccnt)
#define HAS_ASYNC 1
#else
#define HAS_ASYNC 0
#endif

namespace {
constexpr int kNPAIR = 674;          // |i-j|<=5 pairs on 64 nodes
constexpr int kB     = 32;
constexpr int kDIN   = 768;
constexpr int kDHID  = 512;
constexpr int kDOUT  = 256;
constexpr int kDPAIR = 1636;
constexpr int kM     = kB * kNPAIR;  // 21568 rows

constexpr int BM = 128, BN = 128, BK = 16, NT = 8;
constexpr int LDA = 20;   // padded row stride for A tile (floats)
constexpr int LDB = 132;  // padded row stride for B tile (floats)

enum { EPI_BIAS = 0, EPI_GCN = 1, EPI_RELU = 2 };
}

// C[M,N] = A[M,K] @ B[K,N] (+bias, epilogue). Row-major everything.
// 8 wave32s; wave w computes a 16x128 strip: 8 accumulators of 16x16 via
// v_wmma_f32_16x16x4_f32. Double-buffered LDS; tiles staged with
// global_load_async_to_lds_b128 (ASYNCcnt), else sync loads.
__global__ __launch_bounds__(256)
void gemm_wmma_f32(const float* __restrict__ A, const float* __restrict__ Bmat,
                   const float* __restrict__ bias, const float* __restrict__ tvec,
                   float* __restrict__ C, int M, int N, int K, int epi, int npair)
{
    __shared__ float sA[2 * BM * LDA];
    __shared__ float sB[2 * BK * LDB];

    const int tid  = threadIdx.x;
    const int wave = tid >> 5;
    const int lane = tid & 31;
    const int hi   = lane >> 4;   // 0: lanes 0-15, 1: lanes 16-31
    const int l16  = lane & 15;
    const int m0   = blockIdx.y * BM;
    const int n0   = blockIdx.x * BN;
    const int mw   = wave * 16;   // wave's row offset inside block tile

    v8f acc[NT];
#pragma unroll
    for (int i = 0; i < NT; ++i)
#pragma unroll
        for (int j = 0; j < 8; ++j) acc[i][j] = 0.0f;

    // Stage one BK-chunk of A (128x16) and B (16x128) into LDS buffer `buf`.
    auto stage = [&](int ch, int buf) {
        const int k0 = ch * BK;
        float* sAb = &sA[buf * BM * LDA];
        float* sBb = &sB[buf * BK * LDB];
        // A tile: 512 float4 slots, 2 per thread
#pragma unroll
        for (int i = 0; i < 2; ++i) {
            const int slot = tid + i * 256;
            const int row  = slot >> 2;
            const int c4   = (slot & 3) << 2;
            const int gm   = m0 + row;
            const int gk   = k0 + c4;
            float* dst = &sAb[row * LDA + c4];              // 16B aligned
            if (gm < M && gk < K) {                          // K%4==0 => full float4
#if HAS_ASYNC
                __builtin_amdgcn_global_load_async_to_lds_b128(
                    (GLOBAL_AS v4i*)(A + (size_t)gm * K + gk),
                    (LDS_AS v4i*)dst, 0, 0);
#else
                *reinterpret_cast<float4*>(dst) =
                    *reinterpret_cast<const float4*>(A + (size_t)gm * K + gk);
#endif
            } else {
                *reinterpret_cast<float4*>(dst) = make_float4(0.f, 0.f, 0.f, 0.f);
            }
        }
        // B tile: 512 float4 slots, 2 per thread
#pragma unroll
        for (int i = 0; i < 2; ++i) {
            const int slot = tid + i * 256;
            const int row  = slot >> 5;
            const int c4   = (slot & 31) << 2;
            const int gk   = k0 + row;
            float* dst = &sBb[row * LDB + c4];              // 16B aligned
            if (gk < K) {
#if HAS_ASYNC
                __builtin_amdgcn_global_load_async_to_lds_b128(
                    (GLOBAL_AS v4i*)(Bmat + (size_t)gk * N + n0 + c4),
                    (LDS_AS v4i*)dst, 0, 0);
#else
                *reinterpret_cast<float4*>(dst) =
                    *reinterpret_cast<const float4*>(Bmat + (size_t)gk * N + n0 + c4);
#endif
            } else {
                *reinterpret_cast<float4*>(dst) = make_float4(0.f, 0.f, 0.f, 0.f);
            }
        }
    };

    const int nChunks = (K + BK - 1) / BK;   // K is always a multiple of 4
    stage(0, 0);
#if HAS_ASYNC
    __builtin_amdgcn_s_wait_asynccnt(0);
#endif
    __syncthreads();

    for (int ch = 0; ch < nChunks; ++ch) {
        const int cur = ch & 1;
        if (ch + 1 < nChunks) stage(ch + 1, cur ^ 1);   // overlap next-chunk staging

        const float* sAc = &sA[cur * BM * LDA];
        const float* sBc = &sB[cur * BK * LDB];
#pragma unroll
        for (int kk = 0; kk < BK; kk += 4) {
            const int kA = kk + hi * 2;   // ISA 32-bit A layout: V0=K0/K2, V1=K1/K3
            v2f a;
            a.x = sAc[(mw + l16) * LDA + kA];
            a.y = sAc[(mw + l16) * LDA + kA + 1];
#pragma unroll
            for (int nt = 0; nt < NT; ++nt) {
                v2f b;                     // B rows striped across lanes per VGPR
                b.x = sBc[kA * LDB + nt * 16 + l16];
                b.y = sBc[(kA + 1) * LDB + nt * 16 + l16];
                acc[nt] = __builtin_amdgcn_wmma_f32_16x16x4_f32(
                    false, a, false, b, (short)0, acc[nt], false, false);
            }
        }
#if HAS_ASYNC
        __builtin_amdgcn_s_wait_asynccnt(0);
#endif
        __syncthreads();
    }

    // --- epilogue: C/D layout VGPR r -> M = r (+8 for hi lanes), N = lane ---
#pragma unroll
    for (int nt = 0; nt < NT; ++nt) {
        const int col = n0 + nt * 16 + l16;
        const float bb = bias[col];
#pragma unroll
        for (int r = 0; r < 8; ++r) {
            const int row = m0 + mw + r + hi * 8;
            if (row < M) {
                float v = acc[nt][r] + bb;
                if (epi == EPI_GCN) {
                    v = (v + tvec[(row / npair) * N + col]) * 0.5f;
                    v = v > 0.f ? v : 0.f;
                } else if (epi == EPI_RELU) {
                    v = v > 0.f ? v : 0.f;
                }
                C[(size_t)row * N + col] = v;
            }
        }
    }
}

// t[b] = (ac[b] * 674^-1/2) @ W1[2] + b1[2]   -> [32, 512]
__global__ __launch_bounds__(128)
void tvec_kernel(const float* __restrict__ ac, const float* __restrict__ W,
                 const float* __restrict__ bvec, float* __restrict__ t, float scale)
{
    const int n = blockIdx.x * 128 + threadIdx.x;
    const int b = blockIdx.y;
    float s = 0.f;
    for (int k = 0; k < kDIN; ++k)
        s += ac[b * kDIN + k] * W[k * kDHID + n];
    t[b * kDHID + n] = s * scale + bvec[n];
}

// out[row] = hid[row,:] . Wo2[:,0] + bo2[0]; one wave32 per row
__global__ __launch_bounds__(256)
void gemv_out(const float* __restrict__ hid, const float* __restrict__ w,
              const float* __restrict__ b0, float* __restrict__ out, int M)
{
    const int wave = threadIdx.x >> 5;
    const int lane = threadIdx.x & 31;
    const int row  = blockIdx.x * 8 + wave;
    if (row >= M) return;   // wave-uniform
    float s = 0.f;
#pragma unroll
    for (int k = lane; k < kDOUT; k += 32)
        s += hid[(size_t)row * kDOUT + k] * w[k];
#pragma unroll
    for (int o = 16; o > 0; o >>= 1)
        s += __shfl_xor(s, o, 32);
    if (lane == 0) out[row] = s + b0[0];
}

extern "C" void kernel_launch(void* const* d_in, const int* in_sizes, int n_in,
                              void* d_out, int out_size, void* d_ws, size_t ws_size,
                              hipStream_t stream)
{
    (void)in_sizes; (void)n_in; (void)out_size; (void)ws_size;
    const float* cpe = (const float*)d_in[0];
    const float* ac  = (const float*)d_in[3];
    const float* Wi  = (const float*)d_in[4];
    const float* bi  = (const float*)d_in[5];
    const float* W1  = (const float*)d_in[6];
    const float* b1  = (const float*)d_in[7];
    const float* W2  = (const float*)d_in[8];
    const float* b2  = (const float*)d_in[9];
    const float* Wo1 = (const float*)d_in[10];
    const float* bo1 = (const float*)d_in[11];
    const float* Wo2 = (const float*)d_in[12];
    const float* bo2 = (const float*)d_in[13];
    float* out = (float*)d_out;

    // workspace layout (bytes); h2p/hid reuse the dead Xp region
    char* ws = (char*)d_ws;
    float* Xp  = (float*)(ws + 0);             // 21568*768*4  = 66,256,896
    float* hp  = (float*)(ws + 66256896ull);   // 21568*512*4  = 44,171,264
    float* tv  = (float*)(ws + 110428160ull);  // 32*512*4     = 65,536
    float* h2p = (float*)(ws + 0);             // 21568*256*4  (aliases dead Xp)
    float* hid = (float*)(ws + 33554432ull);   // disjoint from h2p, inside Xp region

    const float invs = (float)(1.0 / sqrt((double)kNPAIR));

    tvec_kernel<<<dim3(kDHID / 128, kB), 128, 0, stream>>>(
        ac, W1 + 2 * kDIN * kDHID, b1 + 2 * kDHID, tv, invs);

    const dim3 blk(256);
    const int mb = (kM + BM - 1) / BM;
    // Xp = cpe @ Wi + bi
    gemm_wmma_f32<<<dim3(kDIN / BN, mb), blk, 0, stream>>>(
        cpe, Wi, bi, nullptr, Xp, kM, kDIN, kDPAIR, EPI_BIAS, kNPAIR);
    // h_p = relu((t + Xp @ W1[8] + b1[8]) / 2)
    gemm_wmma_f32<<<dim3(kDHID / BN, mb), blk, 0, stream>>>(
        Xp, W1 + 8 * kDIN * kDHID, b1 + 8 * kDHID, tv, hp, kM, kDHID, kDIN, EPI_GCN, kNPAIR);
    // h2_p = h_p @ W2[8] + b2[8]
    gemm_wmma_f32<<<dim3(kDOUT / BN, mb), blk, 0, stream>>>(
        hp, W2 + 8 * kDHID * kDOUT, b2 + 8 * kDOUT, nullptr, h2p, kM, kDOUT, kDHID, EPI_BIAS, kNPAIR);
    // hid = relu(h2_p @ Wo1 + bo1)
    gemm_wmma_f32<<<dim3(kDOUT / BN, mb), blk, 0, stream>>>(
        h2p, Wo1, bo1, nullptr, hid, kM, kDOUT, kDOUT, EPI_RELU, kNPAIR);
    // out = hid @ Wo2[:,0] + bo2[0]
    gemv_out<<<dim3((kM + 7) / 8), 256, 0, stream>>>(hid, Wo2, bo2, out, kM);
}